// RV2MVImgAttn_75496935129333
// MI455X (gfx1250) — compile-verified
//
#include <hip/hip_runtime.h>
#include <math.h>

#define NQ      8192     // 32*256 queries
#define EMBED   128
#define HEADS   8
#define LEVELS  4
#define POINTS  8
#define NUM_CAMS 6
#define C_RV    256
#define LTOT    30825    // sum of level H*W
#define RMAXF   51.2f

typedef __attribute__((ext_vector_type(2))) float v2f;
typedef __attribute__((ext_vector_type(8))) float v8f;

// ---------------------------------------------------------------------------
// Generic fp32 WMMA GEMM:  C[M x N] = op(A)[M x K] * W^T (W is [N x K]) + bias
// Block = 256 threads = 8 waves; each wave owns a 16x16 tile; block covers
// 16 rows x 128 cols. aColMajor=1 means A stored as [K][M] (in_rv_feat).
// Optional per-row scale (slots/count), residual addMat, ReLU.
// A 16x4 f32 fragment: lanes 0-15 hold M=lane, K={0,1} in v0,v1;
//                      lanes 16-31 hold M=lane-16, K={2,3}.
// B 4x16 fragment mirrors the K split with N=lane&15 striped across lanes.
// C/D: VGPR r, lanes 0-15 -> (M=r, N=lane); lanes 16-31 -> (M=r+8, N=lane-16).
// ---------------------------------------------------------------------------
__global__ void gemm_wmma_kernel(const float* __restrict__ A,
                                 const float* __restrict__ W,
                                 const float* __restrict__ bias,
                                 const float* __restrict__ rowScale,
                                 const float* __restrict__ addMat,
                                 float* __restrict__ C,
                                 int M, int K, int N,
                                 int aColMajor, int applyRelu)
{
    const int tid  = threadIdx.x;
    const int wave = tid >> 5;
    const int lane = tid & 31;
    const int lo   = lane & 15;
    const int hi   = lane >> 4;
    const int p0   = blockIdx.x * 16;
    const int n0   = blockIdx.y * 128 + wave * 16;
    if (n0 >= N) return;                       // wave-uniform; EXEC stays full

    int rowA = p0 + lo;
    if (rowA >= M) rowA = M - 1;               // clamp loads; stores guarded
    const float scale = rowScale ? rowScale[rowA] : 1.0f;
    const float* __restrict__ Wrow = W + (size_t)(n0 + lo) * K;

    v8f c = {};
    if (aColMajor) {
        const float* __restrict__ Acol = A + rowA;
#pragma unroll 4
        for (int k0 = 0; k0 < K; k0 += 4) {
            const int kk = k0 + hi * 2;
            v2f a, b;
            a.x = Acol[(size_t)kk * M];
            a.y = Acol[(size_t)(kk + 1) * M];
            b.x = Wrow[kk];
            b.y = Wrow[kk + 1];
            c = __builtin_amdgcn_wmma_f32_16x16x4_f32(false, a, false, b,
                                                      (short)0, c, false, false);
        }
    } else {
        const float* __restrict__ Arow = A + (size_t)rowA * K;
#pragma unroll 4
        for (int k0 = 0; k0 < K; k0 += 4) {
            const int kk = k0 + hi * 2;
            v2f a, b;
            a.x = Arow[kk]     * scale;
            a.y = Arow[kk + 1] * scale;
            b.x = Wrow[kk];
            b.y = Wrow[kk + 1];
            c = __builtin_amdgcn_wmma_f32_16x16x4_f32(false, a, false, b,
                                                      (short)0, c, false, false);
        }
    }

    const int n = n0 + lo;
    const float bv = bias ? bias[n] : 0.0f;
#pragma unroll
    for (int r = 0; r < 8; ++r) {
        const int m = p0 + r + 8 * hi;
        if (m >= M) continue;
        float val = c[r] + bv;
        if (addMat) val += addMat[(size_t)m * N + n];
        if (applyRelu) val = fmaxf(val, 0.0f);
        C[(size_t)m * N + n] = val;
    }
}

// ---------------------------------------------------------------------------
// Range head: depth_norm = sigmoid(in_rv . w + b); depth = depth_norm * RMAX
// ---------------------------------------------------------------------------
__global__ void range_depth_kernel(const float* __restrict__ in_rv,
                                   const float* __restrict__ rw,
                                   const float* __restrict__ rb,
                                   float* __restrict__ depth,
                                   float* __restrict__ dnorm_out)
{
    int p = blockIdx.x * blockDim.x + threadIdx.x;
    if (p >= NQ) return;
    float acc = rb[0];
#pragma unroll 4
    for (int ch = 0; ch < C_RV; ++ch)
        acc += in_rv[(size_t)ch * NQ + p] * rw[ch];
    float dn = 1.0f / (1.0f + expf(-acc));
    dnorm_out[p] = dn;
    depth[p] = dn * RMAXF;
}

// ---------------------------------------------------------------------------
// Row layernorm over 128 features. One block (128 threads) per row.
// ---------------------------------------------------------------------------
__global__ void layernorm_kernel(const float* __restrict__ in,
                                 float* __restrict__ out,
                                 const float* __restrict__ w,
                                 const float* __restrict__ b)
{
    __shared__ float red[EMBED];
    const int p = blockIdx.x;
    const int t = threadIdx.x;
    float x = in[(size_t)p * EMBED + t];
    red[t] = x;
    __syncthreads();
    for (int s = 64; s > 0; s >>= 1) {
        if (t < s) red[t] += red[t + s];
        __syncthreads();
    }
    float mu = red[0] * (1.0f / EMBED);
    __syncthreads();
    float d = x - mu;
    red[t] = d * d;
    __syncthreads();
    for (int s = 64; s > 0; s >>= 1) {
        if (t < s) red[t] += red[t + s];
        __syncthreads();
    }
    float var = red[0] * (1.0f / EMBED);
    out[(size_t)p * EMBED + t] = d * rsqrtf(var + 1e-5f) * w[t] + b[t];
}

// ---------------------------------------------------------------------------
// In-place softmax over groups of 32 (per query, per head). 8 groups/block.
// ---------------------------------------------------------------------------
__global__ void softmax32_kernel(float* __restrict__ aw)
{
    __shared__ float sm[256];
    const int tid = threadIdx.x;
    const int g   = blockIdx.x * 8 + (tid >> 5);
    const int i   = tid & 31;
    const int sb  = (tid >> 5) * 32;
    const int base = g * 32;
    float x = aw[base + i];
    sm[tid] = x;
    __syncthreads();
    for (int s = 16; s > 0; s >>= 1) {
        if (i < s) sm[tid] = fmaxf(sm[tid], sm[tid + s]);
        __syncthreads();
    }
    float mx = sm[sb];
    __syncthreads();
    float e = expf(x - mx);
    sm[tid] = e;
    __syncthreads();
    for (int s = 16; s > 0; s >>= 1) {
        if (i < s) sm[tid] += sm[tid + s];
        __syncthreads();
    }
    aw[base + i] = e / sm[sb];
}

// ---------------------------------------------------------------------------
// Point sampling: rebuild U_VEC geometry analytically, project by lidar2img.
// One thread per (cam, query). Writes normalized (x,y) ref and mask.
// ---------------------------------------------------------------------------
__global__ void point_sampling_kernel(const float* __restrict__ depth,
                                      const float* __restrict__ l2i,
                                      const float* __restrict__ img_shapes,
                                      float* __restrict__ refxy,
                                      float* __restrict__ maskb)
{
    int t = blockIdx.x * blockDim.x + threadIdx.x;
    if (t >= NUM_CAMS * NQ) return;
    const int cam = t / NQ;
    const int p   = t - cam * NQ;
    const int hh  = p >> 8;   // row (32)
    const int ww  = p & 255;  // col (256)

    const float PI = 3.14159265358979323846f;
    // az_line = linspace(-pi,pi,1025)[:-1] + step/2; az[w] = mean of 4 -> closed form
    const float step = PI / 512.0f;
    const float az   = -PI + (4.0f * (float)ww + 2.0f) * step;
    // ELEV_DEG = linspace(10.67,-30.67,32), reversed per row
    const float elev = (10.67f - (float)(31 - hh) * (41.34f / 31.0f)) * (PI / 180.0f);
    const float ce = cosf(elev), se = sinf(elev);
    const float ux = cosf(az) * ce, uy = sinf(az) * ce, uz = se;

    const float d = depth[p];
    const float X = d * ux, Y = d * uy, Z = d * uz;
    const float* __restrict__ M = l2i + cam * 16;
    const float r0 = M[0] * X + M[1] * Y + M[2]  * Z + M[3];
    const float r1 = M[4] * X + M[5] * Y + M[6]  * Z + M[7];
    const float r2 = M[8] * X + M[9] * Y + M[10] * Z + M[11];

    const float eps = 1e-5f;
    const float den = fmaxf(r2, eps);
    const float x = (r0 / den) / img_shapes[1];   // /1600 (W)
    const float y = (r1 / den) / img_shapes[0];   // /928  (H)
    const int valid = (r2 > eps) && (y > 0.0f) && (y < 1.0f) &&
                      (x < 1.0f) && (x > 0.0f);
    refxy[(size_t)t * 2 + 0] = x;
    refxy[(size_t)t * 2 + 1] = y;
    maskb[t] = valid ? 1.0f : 0.0f;
}

__global__ void count_kernel(const float* __restrict__ maskb,
                             float* __restrict__ invc)
{
    int p = blockIdx.x * blockDim.x + threadIdx.x;
    if (p >= NQ) return;
    float s = 0.0f;
    for (int c = 0; c < NUM_CAMS; ++c) s += maskb[c * NQ + p];
    invc[p] = 1.0f / fmaxf(s, 1.0f);
}

// ---------------------------------------------------------------------------
// Deformable sampling + attention. 16-lane group per (q, head); d = lane&15.
// Loops cams internally (masked sum) -> plain store, no atomics/zero-init.
// All gathers of v are 16 consecutive floats across the group (64B, L2-hot).
// ---------------------------------------------------------------------------
__device__ __forceinline__ float tapv(const float* __restrict__ vcam,
                                      int st, int Wl, int Hl,
                                      int xi, int yi, int hd, float wgt)
{
    if (xi < 0 || xi >= Wl || yi < 0 || yi >= Hl) return 0.0f;
    return wgt * vcam[(size_t)(st + yi * Wl + xi) * EMBED + hd];
}

__global__ void sample_attn_kernel(const float* __restrict__ v,
                                   const float* __restrict__ off,
                                   const float* __restrict__ aw,
                                   const float* __restrict__ refxy,
                                   const float* __restrict__ maskb,
                                   float* __restrict__ slots)
{
    const int LH[LEVELS] = {116, 58, 29, 15};
    const int LW[LEVELS] = {200, 100, 50, 25};
    const int LS[LEVELS] = {0, 23200, 29000, 30450};

    const int tid  = threadIdx.x;
    const int gidx = blockIdx.x * 16 + (tid >> 4);
    const int d    = tid & 15;
    const int q    = gidx >> 3;
    const int h    = gidx & 7;
    if (q >= NQ) return;
    const int hd = h * 16 + d;

    float acc = 0.0f;
    for (int cam = 0; cam < NUM_CAMS; ++cam) {
        const float valid = maskb[cam * NQ + q];
        if (valid == 0.0f) continue;          // out*valid == 0 anyway
        const float rx = refxy[(size_t)(cam * NQ + q) * 2 + 0];
        const float ry = refxy[(size_t)(cam * NQ + q) * 2 + 1];
        const float* __restrict__ vcam = v + (size_t)cam * LTOT * EMBED;
        float acam = 0.0f;
#pragma unroll
        for (int l = 0; l < LEVELS; ++l) {
            const int Hl = LH[l], Wl = LW[l], st = LS[l];
            const float fW = (float)Wl, fH = (float)Hl;
#pragma unroll
            for (int pt = 0; pt < POINTS; ++pt) {
                const int oi = q * 512 + (((h * 4 + l) * 8 + pt) * 2);
                const float lx = rx + off[oi]     / fW;
                const float ly = ry + off[oi + 1] / fH;
                // grid->pixel: x = ((2*loc-1)+1)*W/2 - 0.5 = loc*W - 0.5
                const float xs = lx * fW - 0.5f;
                const float ys = ly * fH - 0.5f;
                const float x0f = floorf(xs), y0f = floorf(ys);
                const float wx1 = xs - x0f, wy1 = ys - y0f;
                const int x0 = (int)x0f, y0 = (int)y0f;
                const float aww = aw[q * 256 + h * 32 + l * 8 + pt];
                float s = 0.0f;
                s += tapv(vcam, st, Wl, Hl, x0,     y0,     hd, (1.0f - wx1) * (1.0f - wy1));
                s += tapv(vcam, st, Wl, Hl, x0 + 1, y0,     hd, wx1 * (1.0f - wy1));
                s += tapv(vcam, st, Wl, Hl, x0,     y0 + 1, hd, (1.0f - wx1) * wy1);
                s += tapv(vcam, st, Wl, Hl, x0 + 1, y0 + 1, hd, wx1 * wy1);
                acam += aww * s;
            }
        }
        acc += valid * acam;
    }
    slots[(size_t)q * EMBED + hd] = acc;
}

// ---------------------------------------------------------------------------
extern "C" void kernel_launch(void* const* d_in, const int* in_sizes, int n_in,
                              void* d_out, int out_size, void* d_ws, size_t ws_size,
                              hipStream_t stream)
{
    (void)in_sizes; (void)n_in; (void)out_size; (void)ws_size;

    const float* in_rv   = (const float*)d_in[0];
    const float* mlvl    = (const float*)d_in[1];   // [6][30825][1][128] row-major
    const float* l2i     = (const float*)d_in[2];
    const float* imgsh   = (const float*)d_in[3];
    const float* projq_w = (const float*)d_in[4];
    const float* projq_b = (const float*)d_in[5];
    const float* rh_w    = (const float*)d_in[6];
    const float* rh_b    = (const float*)d_in[7];
    const float* n1_w    = (const float*)d_in[8];
    const float* n1_b    = (const float*)d_in[9];
    const float* vp_w    = (const float*)d_in[10];
    const float* vp_b    = (const float*)d_in[11];
    const float* off_w   = (const float*)d_in[12];
    const float* off_b   = (const float*)d_in[13];
    const float* aw_w    = (const float*)d_in[14];
    const float* aw_b    = (const float*)d_in[15];
    const float* op_w    = (const float*)d_in[16];
    const float* op_b    = (const float*)d_in[17];
    const float* n2_w    = (const float*)d_in[18];
    const float* n2_b    = (const float*)d_in[19];
    const float* f1_w    = (const float*)d_in[20];
    const float* f1_b    = (const float*)d_in[21];
    const float* f2_w    = (const float*)d_in[22];
    const float* f2_b    = (const float*)d_in[23];

    float* out       = (float*)d_out;             // fused flat [q][o] (reshape!)
    float* dnorm_out = out + (size_t)EMBED * NQ;  // depth_norm tail (8192)

    float* ws = (float*)d_ws;
    float* v      = ws; ws += (size_t)NUM_CAMS * LTOT * EMBED;  // 94.7 MB
    float* qraw   = ws; ws += (size_t)NQ * EMBED;
    float* qln    = ws; ws += (size_t)NQ * EMBED;
    float* offb   = ws; ws += (size_t)NQ * 512;
    float* awb    = ws; ws += (size_t)NQ * 256;
    float* depth  = ws; ws += NQ;
    float* refxy  = ws; ws += (size_t)NUM_CAMS * NQ * 2;
    float* maskb  = ws; ws += (size_t)NUM_CAMS * NQ;
    float* invc   = ws; ws += NQ;
    float* slots  = ws; ws += (size_t)NQ * EMBED;
    float* fused  = ws; ws += (size_t)NQ * EMBED;
    float* normed = ws; ws += (size_t)NQ * EMBED;
    float* h1     = ws; ws += (size_t)NQ * EMBED;

    const int MV = NUM_CAMS * LTOT;               // 184950 value rows

    // 1. depth head (also writes depth_norm output)
    range_depth_kernel<<<NQ / 256, 256, 0, stream>>>(in_rv, rh_w, rh_b, depth, dnorm_out);
    // 2. proj_q: q_raw[q][o], A column-major ([C][HW]), K=256
    gemm_wmma_kernel<<<dim3(NQ / 16, 1), 256, 0, stream>>>(
        in_rv, projq_w, projq_b, nullptr, nullptr, qraw, NQ, C_RV, EMBED, 1, 0);
    // 3. layernorm1 -> query
    layernorm_kernel<<<NQ, EMBED, 0, stream>>>(qraw, qln, n1_w, n1_b);
    // 4. vproj over all cams/levels: [184950x128]*[128x128]
    gemm_wmma_kernel<<<dim3((MV + 15) / 16, 1), 256, 0, stream>>>(
        mlvl, vp_w, vp_b, nullptr, nullptr, v, MV, EMBED, EMBED, 0, 0);
    // 5. sampling offsets (cam-independent): N=512
    gemm_wmma_kernel<<<dim3(NQ / 16, 4), 256, 0, stream>>>(
        qln, off_w, off_b, nullptr, nullptr, offb, NQ, EMBED, 512, 0, 0);
    // 6. attention logits: N=256
    gemm_wmma_kernel<<<dim3(NQ / 16, 2), 256, 0, stream>>>(
        qln, aw_w, aw_b, nullptr, nullptr, awb, NQ, EMBED, 256, 0, 0);
    // 7. softmax over 32 (level*point) per (q, head)
    softmax32_kernel<<<(NQ * HEADS) / 8, 256, 0, stream>>>(awb);
    // 8. project range points to each camera
    point_sampling_kernel<<<(NUM_CAMS * NQ + 255) / 256, 256, 0, stream>>>(
        depth, l2i, imgsh, refxy, maskb);
    // 9. 1/max(count,1)
    count_kernel<<<NQ / 256, 256, 0, stream>>>(maskb, invc);
    // 10. deformable sampling + head attention, masked cam sum -> slots
    sample_attn_kernel<<<(NQ * HEADS) / 16, 256, 0, stream>>>(
        v, offb, awb, refxy, maskb, slots);
    // 11. output proj: fused = (slots/count) @ op_w^T + op_b + query
    gemm_wmma_kernel<<<dim3(NQ / 16, 1), 256, 0, stream>>>(
        slots, op_w, op_b, invc, qln, fused, NQ, EMBED, EMBED, 0, 0);
    // 12. layernorm2
    layernorm_kernel<<<NQ, EMBED, 0, stream>>>(fused, normed, n2_w, n2_b);
    // 13. ffn1 + ReLU
    gemm_wmma_kernel<<<dim3(NQ / 16, 1), 256, 0, stream>>>(
        normed, f1_w, f1_b, nullptr, nullptr, h1, NQ, EMBED, EMBED, 0, 1);
    // 14. ffn2 + residual, straight into d_out (reference reshape == flat [q][o])
    gemm_wmma_kernel<<<dim3(NQ / 16, 1), 256, 0, stream>>>(
        h1, f2_w, f2_b, nullptr, fused, out, NQ, EMBED, EMBED, 0, 0);
}